// SymplecticPropagator_64355789964150
// MI455X (gfx1250) — compile-verified
//
#include <hip/hip_runtime.h>
#include <hip/hip_bf16.h>
#include <math.h>
#include <stdint.h>

// ---------------------------------------------------------------------------
// SymplecticPropagator for MI455X (gfx1250, wave32).
//   A = triu(K,1) - triu(K,1)^T            (real skew-symmetric)
//   S = A^T A = -A*A                        (WMMA f32 GEMM, async-LDS staged)
//   S = W diag(theta^2) W^T                 (blocked parallel Jacobi;
//                                            64x64 subproblems in LDS,
//                                            block-rotation applies via WMMA)
//   U = A*W                                 (WMMA f32 GEMM)
//   eigenpairs of H=iA:  v(+/-) = (w +/- i*Aw/theta)/sqrt(2), eig = +/-theta
//   evo = cos(lambda*dt) - i*sin(lambda*dt)
// ---------------------------------------------------------------------------

#define DIMN 2048
#define NBLK 64          // number of 32-wide column blocks
#define PAIRS 32         // NBLK/2 independent block pairs per round
#define ROUNDS 63        // NBLK-1 rounds per sweep
#define OUTER_SWEEPS 8
#define INNER_SWEEPS 8
#define KC 32            // GEMM k-chunk staged in LDS

typedef __attribute__((ext_vector_type(2))) float v2f;
typedef __attribute__((ext_vector_type(8))) float v8f;

__device__ __forceinline__ v8f wmma_f32(v2f a, v2f b, v8f c) {
  // V_WMMA_F32_16X16X4_F32 : D(16x16,f32) = A(16x4,f32)*B(4x16,f32) + C
  return __builtin_amdgcn_wmma_f32_16x16x4_f32(
      /*neg_a=*/false, a, /*neg_b=*/false, b,
      /*c_mod=*/(short)0, c, /*reuse_a=*/false, /*reuse_b=*/false);
}

// ASYNCcnt-tracked global->LDS copy, 16B per lane (CDNA5 async memory path).
__device__ __forceinline__ void async_g2l_b128(unsigned lds_off, const float* gaddr) {
  asm volatile("global_load_async_to_lds_b128 %0, %1, off"
               :: "v"(lds_off), "v"(gaddr) : "memory");
}
__device__ __forceinline__ void wait_asynccnt0() {
  asm volatile("s_wait_asynccnt 0x0" ::: "memory");
}

// Round-robin tournament pairing: n even, rounds in [0,n-1), pair i in [0,n/2).
__device__ __forceinline__ void rr_pair(int n, int round, int i, int* p, int* q) {
  int m = n - 1;
  int a, b;
  if (i == 0) { a = m; b = round % m; }
  else { a = (round + i) % m; b = (round - i + 2 * m) % m; }
  *p = (a < b) ? a : b;
  *q = (a < b) ? b : a;
}

__device__ __forceinline__ int blk_map(int bp, int bq, int localIdx) {
  return (localIdx < 32) ? (bp * 32 + localIdx) : (bq * 32 + (localIdx - 32));
}

// --------------------------------------------------------------------------
__global__ void build_skew(const float* __restrict__ K, float* __restrict__ A, int n) {
  int idx = blockIdx.x * blockDim.x + threadIdx.x;
  if (idx >= n * n) return;
  int r = idx / n, c = idx - r * n;
  float up = (c > r) ? K[r * n + c] : 0.0f;
  float lo = (r > c) ? K[c * n + r] : 0.0f;
  A[idx] = up - lo;
}

__global__ void set_identity(float* __restrict__ W, int n) {
  int idx = blockIdx.x * blockDim.x + threadIdx.x;
  if (idx >= n * n) return;
  int r = idx / n, c = idx - r * n;
  W[idx] = (r == c) ? 1.0f : 0.0f;
}

// C = alpha * X * Y.  256 threads = 8 waves per 64x64 output tile; two 16x16
// WMMA accumulators per wave.  K-chunks of 32 double-buffered in LDS via
// GLOBAL_LOAD_ASYNC_TO_LDS_B128 (ASYNCcnt), so the next chunk's DMA overlaps
// the current chunk's v_wmma_f32_16x16x4_f32 work.
__global__ __launch_bounds__(256) void gemm_nn_wmma(
    const float* __restrict__ X, const float* __restrict__ Y,
    float* __restrict__ C, int n, float alpha) {
  // pads keep rows 16B-aligned (36*4=144=9*16, 68*4=272=17*16) and make the
  // strided A-column reads land on 16 distinct LDS banks.
  __shared__ float Al[2][64][36];
  __shared__ float Bl[2][32][68];

  int tid = threadIdx.x;
  int r0 = blockIdx.y << 6;         // output row base
  int c0 = blockIdx.x << 6;         // output col base
  int lane = tid & 31, wave = tid >> 5;
  int l = lane & 15, half = lane >> 4;
  int rt = wave & 3;                // row tile 0..3
  int ct0 = (wave >> 2) * 2;        // col tiles ct0, ct0+1

  v8f acc0 = {}, acc1 = {};

  // ---- stage chunk `kc0` into buffer `buf` (async, no wait here) ----
  auto stage = [&](int buf, int kc0) {
    // A: 64 rows x 32 k  -> 512 x 16B items
    for (int it = tid; it < 512; it += 256) {
      int r = it >> 3, c4 = (it & 7) << 2;
      unsigned lds = (unsigned)(uintptr_t)&Al[buf][r][c4];
      async_g2l_b128(lds, &X[(size_t)(r0 + r) * n + kc0 + c4]);
    }
    // B: 32 k x 64 cols -> 512 x 16B items
    for (int it = tid; it < 512; it += 256) {
      int k = it >> 4, c4 = (it & 15) << 2;
      unsigned lds = (unsigned)(uintptr_t)&Bl[buf][k][c4];
      async_g2l_b128(lds, &Y[(size_t)(kc0 + k) * n + c0 + c4]);
    }
  };

  stage(0, 0);
  wait_asynccnt0();
  __syncthreads();

  const int nchunks = n / KC;       // 64
  for (int ch = 0; ch < nchunks; ++ch) {
    int cur = ch & 1;
    if (ch + 1 < nchunks) stage(cur ^ 1, (ch + 1) * KC);

#pragma unroll
    for (int k0 = 0; k0 < KC; k0 += 4) {
      int ka = k0 + half * 2;
      v2f a, b0, b1;
      a.x = Al[cur][rt * 16 + l][ka];
      a.y = Al[cur][rt * 16 + l][ka + 1];
      b0.x = Bl[cur][ka][ct0 * 16 + l];
      b0.y = Bl[cur][ka + 1][ct0 * 16 + l];
      b1.x = Bl[cur][ka][ct0 * 16 + 16 + l];
      b1.y = Bl[cur][ka + 1][ct0 * 16 + 16 + l];
      acc0 = wmma_f32(a, b0, acc0);
      acc1 = wmma_f32(a, b1, acc1);
    }
    wait_asynccnt0();   // next chunk's async DMA has landed
    __syncthreads();    // everyone done reading `cur` before it is re-staged
  }

#pragma unroll
  for (int r = 0; r < 8; ++r) {
    int row = r0 + rt * 16 + r + half * 8;
    int col = c0 + ct0 * 16 + l;
    C[(size_t)row * n + col]      = alpha * acc0[r];
    C[(size_t)row * n + col + 16] = alpha * acc1[r];
  }
}

// One workgroup per block-pair: diagonalize the 64x64 subproblem in LDS with
// cyclic scalar Jacobi, accumulate the 64x64 rotation R, store to Rbuf.
__global__ void block_jacobi_solve(const float* __restrict__ S, float* __restrict__ Rbuf,
                                   int step, int n) {
  __shared__ float Sh[64][65];
  __shared__ float Rl[64][65];
  __shared__ float cc[32], ssn[32];
  __shared__ int pr[32], qr[32];
  int tid = threadIdx.x;           // 256 threads
  int pairId = blockIdx.x;
  int bp, bq;
  rr_pair(NBLK, step, pairId, &bp, &bq);
  for (int idx = tid; idx < 4096; idx += 256) {
    int i = idx >> 6, j = idx & 63;
    Sh[i][j] = S[blk_map(bp, bq, i) * n + blk_map(bp, bq, j)];
    Rl[i][j] = (i == j) ? 1.0f : 0.0f;
  }
  __syncthreads();
  for (int sw = 0; sw < INNER_SWEEPS; ++sw) {
    for (int rd = 0; rd < 63; ++rd) {
      if (tid < 32) {
        int p, q;
        rr_pair(64, rd, tid, &p, &q);
        float app = Sh[p][p], aqq = Sh[q][q], apq = Sh[p][q];
        float c = 1.0f, s = 0.0f;
        if (fabsf(apq) > 1e-30f) {
          float tau = (aqq - app) / (2.0f * apq);
          float t = (tau >= 0.0f ? 1.0f : -1.0f) /
                    (fabsf(tau) + sqrtf(1.0f + tau * tau));
          c = rsqrtf(1.0f + t * t);
          s = t * c;
        }
        pr[tid] = p; qr[tid] = q; cc[tid] = c; ssn[tid] = s;
      }
      __syncthreads();
      // right-apply to columns of Sh and Rl (pairs touch disjoint columns)
      for (int idx = tid; idx < 2048; idx += 256) {
        int pi = idx & 31, r = idx >> 5;
        int p = pr[pi], q = qr[pi];
        float c = cc[pi], s = ssn[pi];
        float x = Sh[r][p], y = Sh[r][q];
        Sh[r][p] = c * x - s * y;
        Sh[r][q] = s * x + c * y;
        x = Rl[r][p]; y = Rl[r][q];
        Rl[r][p] = c * x - s * y;
        Rl[r][q] = s * x + c * y;
      }
      __syncthreads();
      // left-apply to rows of Sh
      for (int idx = tid; idx < 2048; idx += 256) {
        int pi = idx & 31, col = idx >> 5;
        int p = pr[pi], q = qr[pi];
        float c = cc[pi], s = ssn[pi];
        float x = Sh[p][col], y = Sh[q][col];
        Sh[p][col] = c * x - s * y;
        Sh[q][col] = s * x + c * y;
      }
      __syncthreads();
    }
  }
  float* Rp = Rbuf + pairId * 4096;
  for (int idx = tid; idx < 4096; idx += 256) Rp[idx] = Rl[idx >> 6][idx & 63];
}

// C[:, I] <- C[:, I] * R for each pair (disjoint column sets).  WMMA f32.
__global__ void apply_right(float* __restrict__ C, const float* __restrict__ Rbuf,
                            int step, int n) {
  __shared__ float Rl[64][64];
  __shared__ float Xs[16][65];
  int tid = threadIdx.x;           // 128 threads = 4 waves
  int pairId = blockIdx.y;
  int r0 = blockIdx.x << 4;
  int bp, bq;
  rr_pair(NBLK, step, pairId, &bp, &bq);
  const float* Rp = Rbuf + pairId * 4096;
  for (int idx = tid; idx < 4096; idx += 128) Rl[idx >> 6][idx & 63] = Rp[idx];
  for (int idx = tid; idx < 1024; idx += 128) {
    int r = idx >> 6, cL = idx & 63;
    Xs[r][cL] = C[(r0 + r) * n + blk_map(bp, bq, cL)];
  }
  __syncthreads();
  int lane = tid & 31, wave = tid >> 5;
  int l = lane & 15, half = lane >> 4;
  v8f acc = {};
#pragma unroll
  for (int k0 = 0; k0 < 64; k0 += 4) {
    int ka = k0 + half * 2;
    v2f a, b;
    a.x = Xs[l][ka];
    a.y = Xs[l][ka + 1];
    int cL = wave * 16 + l;
    b.x = Rl[ka][cL];
    b.y = Rl[ka + 1][cL];
    acc = wmma_f32(a, b, acc);
  }
#pragma unroll
  for (int r = 0; r < 8; ++r) {
    int row = r0 + r + half * 8;
    int cL = wave * 16 + l;
    C[row * n + blk_map(bp, bq, cL)] = acc[r];
  }
}

// C[I, :] <- R^T * C[I, :] for each pair (disjoint row sets).  WMMA f32.
__global__ void apply_left(float* __restrict__ C, const float* __restrict__ Rbuf,
                           int step, int n) {
  __shared__ float Rl[64][64];
  __shared__ float Xs[64][17];
  int tid = threadIdx.x;           // 128 threads = 4 waves
  int pairId = blockIdx.y;
  int c0 = blockIdx.x << 4;
  int bp, bq;
  rr_pair(NBLK, step, pairId, &bp, &bq);
  const float* Rp = Rbuf + pairId * 4096;
  for (int idx = tid; idx < 4096; idx += 128) Rl[idx >> 6][idx & 63] = Rp[idx];
  for (int idx = tid; idx < 1024; idx += 128) {
    int k = idx >> 4, cL = idx & 15;
    Xs[k][cL] = C[blk_map(bp, bq, k) * n + c0 + cL];
  }
  __syncthreads();
  int lane = tid & 31, wave = tid >> 5;
  int l = lane & 15, half = lane >> 4;
  v8f acc = {};
#pragma unroll
  for (int k0 = 0; k0 < 64; k0 += 4) {
    int ka = k0 + half * 2;
    v2f a, b;
    a.x = Rl[ka][wave * 16 + l];        // A[m][k] = R[k][m]  (R^T)
    a.y = Rl[ka + 1][wave * 16 + l];
    b.x = Xs[ka][l];
    b.y = Xs[ka + 1][l];
    acc = wmma_f32(a, b, acc);
  }
#pragma unroll
  for (int r = 0; r < 8; ++r) {
    int mL = wave * 16 + r + half * 8;
    C[blk_map(bp, bq, mL) * n + c0 + l] = acc[r];
  }
}

__global__ void extract_theta(const float* __restrict__ S, float* __restrict__ theta, int n) {
  int k = blockIdx.x * blockDim.x + threadIdx.x;
  if (k >= n) return;
  theta[k] = sqrtf(fmaxf(S[k * n + k], 0.0f));
}

// Single-block bitonic argsort of 2048 values, ascending.
__global__ void bitonic_argsort(const float* __restrict__ th,
                                float* __restrict__ sv, int* __restrict__ si) {
  __shared__ float v[2048];
  __shared__ int ix[2048];
  int tid = threadIdx.x;           // 1024 threads
  for (int i = tid; i < 2048; i += 1024) { v[i] = th[i]; ix[i] = i; }
  __syncthreads();
  for (int k = 2; k <= 2048; k <<= 1) {
    for (int j = k >> 1; j > 0; j >>= 1) {
      for (int i = tid; i < 2048; i += 1024) {
        int partner = i ^ j;
        if (partner > i) {
          bool up = ((i & k) == 0);
          float a = v[i], b = v[partner];
          if ((a > b) == up) {
            v[i] = b; v[partner] = a;
            int t = ix[i]; ix[i] = ix[partner]; ix[partner] = t;
          }
        }
      }
      __syncthreads();
    }
  }
  for (int i = tid; i < 2048; i += 1024) { sv[i] = v[i]; si[i] = ix[i]; }
}

// Pair consecutive sorted eigenvalues (doubly degenerate), build lambda array
// in eigh's ascending order: [-th_{1023} ... -th_0, +th_0 ... +th_{1023}].
__global__ void build_pairs(const float* __restrict__ sv, const int* __restrict__ si,
                            float* __restrict__ lambda, float* __restrict__ pth,
                            int* __restrict__ pwi) {
  int t = blockIdx.x * blockDim.x + threadIdx.x;
  if (t >= 1024) return;
  float thv = 0.5f * (sv[2 * t] + sv[2 * t + 1]);
  pth[t] = thv;
  pwi[t] = si[2 * t];
  lambda[1023 - t] = -thv;
  lambda[1024 + t] = thv;
}

// V[r, col]: column col<1024 -> pair t=1023-col, eig -theta; else +theta.
// v(+/-) = (w +/- i*u/theta)/sqrt(2) with u = A*w.
__global__ void emit_V(const float* __restrict__ W, const float* __restrict__ U,
                       const float* __restrict__ pth, const int* __restrict__ pwi,
                       float* __restrict__ V, int n) {
  int r = blockIdx.x;
  for (int col = threadIdx.x; col < n; col += blockDim.x) {
    int t, sgn;
    if (col < n / 2) { t = n / 2 - 1 - col; sgn = -1; }
    else             { t = col - n / 2;     sgn =  1; }
    int wi = pwi[t];
    float th = fmaxf(pth[t], 1e-30f);
    float wre = W[r * n + wi] * 0.70710678118654752f;
    float uim = U[r * n + wi] * ((float)sgn * 0.70710678118654752f / th);
    V[(size_t)(r * n + col) * 2]     = wre;
    V[(size_t)(r * n + col) * 2 + 1] = uim;
  }
}

// V_inv = conj(V)^T : Vinv[cv, r] = conj(V[r, cv]).
__global__ void emit_Vinv(const float* __restrict__ W, const float* __restrict__ U,
                          const float* __restrict__ pth, const int* __restrict__ pwi,
                          float* __restrict__ Vinv, int n) {
  int cv = blockIdx.x;   // row index of V_inv == column index of V
  int t, sgn;
  if (cv < n / 2) { t = n / 2 - 1 - cv; sgn = -1; }
  else            { t = cv - n / 2;     sgn =  1; }
  int wi = pwi[t];
  float th = fmaxf(pth[t], 1e-30f);
  float sc = (float)sgn * 0.70710678118654752f / th;
  for (int r = threadIdx.x; r < n; r += blockDim.x) {
    float wre = W[r * n + wi] * 0.70710678118654752f;
    float uim = U[r * n + wi] * sc;
    Vinv[(size_t)(cv * n + r) * 2]     = wre;
    Vinv[(size_t)(cv * n + r) * 2 + 1] = -uim;
  }
}

// evo[b,t,k] = exp(-i*lambda_k*dt[b,t]) = cos - i*sin
__global__ void emit_evo(const float* __restrict__ lambda, const float* __restrict__ dt,
                         float* __restrict__ evo, int n) {
  int bt = blockIdx.x;   // 0..2047 (8*256 flattened)
  float d = dt[bt];
  for (int k = threadIdx.x; k < n; k += blockDim.x) {
    float ph = lambda[k] * d;
    float s, c;
    __sincosf(ph, &s, &c);
    evo[(size_t)(bt * n + k) * 2]     = c;
    evo[(size_t)(bt * n + k) * 2 + 1] = -s;
  }
}

// ---------------------------------------------------------------------------
extern "C" void kernel_launch(void* const* d_in, const int* in_sizes, int n_in,
                              void* d_out, int out_size, void* d_ws, size_t ws_size,
                              hipStream_t stream) {
  (void)in_sizes; (void)n_in; (void)out_size; (void)ws_size;
  const int N = DIMN;
  const size_t NN = (size_t)N * N;            // 4,194,304

  const float* K  = (const float*)d_in[0];
  const float* dt = (const float*)d_in[1];    // 8*256 = 2048 values

  float* fws   = (float*)d_ws;
  float* A     = fws;
  float* S     = fws + NN;
  float* W     = fws + 2 * NN;
  float* U     = fws + 3 * NN;
  float* Rbuf  = fws + 4 * NN;                // 32 * 64*64 = 131072 floats
  float* theta = Rbuf + 131072;
  float* sval  = theta + 2048;
  int*   sidx  = (int*)(sval + 2048);
  float* lam   = (float*)(sidx + 2048);
  float* pth   = lam + 2048;
  int*   pwi   = (int*)(pth + 1024);

  float* Vout    = (float*)d_out;             // 2048*2048 complex
  float* Vinvout = Vout + 2 * NN;
  float* evoout  = Vout + 4 * NN;             // (8,256,2048) complex

  // A = triu(K,1) - triu(K,1)^T
  build_skew<<<(NN + 255) / 256, 256, 0, stream>>>(K, A, N);
  // S = A^T A = -A*A   (WMMA, async-LDS double buffered)
  gemm_nn_wmma<<<dim3(N / 64, N / 64), 256, 0, stream>>>(A, A, S, N, -1.0f);
  // W = I
  set_identity<<<(NN + 255) / 256, 256, 0, stream>>>(W, N);

  // Blocked parallel cyclic Jacobi on S, accumulating W.
  for (int sweep = 0; sweep < OUTER_SWEEPS; ++sweep) {
    for (int rd = 0; rd < ROUNDS; ++rd) {
      block_jacobi_solve<<<PAIRS, 256, 0, stream>>>(S, Rbuf, rd, N);
      apply_right<<<dim3(N / 16, PAIRS), 128, 0, stream>>>(S, Rbuf, rd, N);
      apply_left <<<dim3(N / 16, PAIRS), 128, 0, stream>>>(S, Rbuf, rd, N);
      apply_right<<<dim3(N / 16, PAIRS), 128, 0, stream>>>(W, Rbuf, rd, N);
    }
  }

  // U = A*W   (WMMA)
  gemm_nn_wmma<<<dim3(N / 64, N / 64), 256, 0, stream>>>(A, W, U, N, 1.0f);

  extract_theta<<<(N + 255) / 256, 256, 0, stream>>>(S, theta, N);
  bitonic_argsort<<<1, 1024, 0, stream>>>(theta, sval, sidx);
  build_pairs<<<4, 256, 0, stream>>>(sval, sidx, lam, pth, pwi);

  emit_V   <<<N, 256, 0, stream>>>(W, U, pth, pwi, Vout, N);
  emit_Vinv<<<N, 256, 0, stream>>>(W, U, pth, pwi, Vinvout, N);
  emit_evo <<<8 * 256, 256, 0, stream>>>(lam, dt, evoout, N);
}